// MultiHeadAttention_11330123727139
// MI455X (gfx1250) — compile-verified
//
#include <hip/hip_runtime.h>
#include <math.h>

// ---------------------------------------------------------------------------
// MultiHeadAttention, algebraically refactored for n=1:
//   q[h,:]  = Q·WQ[h]^T + bQ[h]                     (8 GEMVs, reads 32MB)
//   qw[h,:] = q[h,:]·WK[h]   (push q through WK)    (8 GEMVs, reads 32MB)
//   c[h]    = q[h,:]·bK[h,:]
//   logits[h,m] = (qw[h,:]·K[m,:] + c[h]) / 1024    (reads 8MB)
//   w[h,:]  = softmax_m(logits[h,:])
//   out[h,p]= sum_m w[h,m] * V[m,p]                 (WMMA f32 16x16x4, reads 400MB)
// Total ~475MB HBM traffic, ~1.7 GFLOP -> bandwidth bound (~20us @ 23.3TB/s).
// V-stream kernel: 2 N-tiles/wave + K-split(2) => ~3072 waves x ~40 loads in
// flight ~= 16MB of MLP, enough to saturate HBM.
// ---------------------------------------------------------------------------

typedef __attribute__((ext_vector_type(2))) float v2f;
typedef __attribute__((ext_vector_type(4))) float v4f;
typedef __attribute__((ext_vector_type(8))) float v8f;

#define DIM   1024
#define MNB   2048
#define NH    8
#define NPIX  49152   // 128*128*3
#define KSPLIT 2
#define KCHUNK (MNB / KSPLIT)

// ---- q[h,e] = dot(Q, WQ[h,e,:]) + bQ[h,e] ; one wave per output -------------
__global__ void proj_q_kernel(const float* __restrict__ Q,
                              const float* __restrict__ WQ,
                              const float* __restrict__ bQ,
                              float* __restrict__ q) {
  int wid  = (blockIdx.x * blockDim.x + threadIdx.x) >> 5;  // h*1024+e
  int lane = threadIdx.x & 31;
  const float* w = WQ + (size_t)wid * DIM;
  float s = 0.f;
  for (int d = lane; d < DIM; d += 32) s = fmaf(Q[d], w[d], s);
  for (int off = 16; off; off >>= 1) s += __shfl_xor(s, off, 32);
  if (lane == 0) q[wid] = s + bQ[wid];
}

// ---- qw[h,d] = sum_e q[h,e]*WK[h,e,d] ; thread per (h,d), coalesced in d ---
__global__ void proj_qw_kernel(const float* __restrict__ q,
                               const float* __restrict__ WK,
                               float* __restrict__ qw) {
  int idx = blockIdx.x * blockDim.x + threadIdx.x;  // h*1024+d
  int h = idx >> 10;
  int d = idx & 1023;
  const float* w  = WK + (size_t)h * DIM * DIM + d;
  const float* qh = q + h * DIM;
  float s = 0.f;
  for (int e = 0; e < DIM; ++e) s = fmaf(qh[e], w[(size_t)e * DIM], s);
  qw[idx] = s;
}

// ---- c[h] = dot(q[h,:], bK[h,:]) ; one wave per head -----------------------
__global__ void proj_c_kernel(const float* __restrict__ q,
                              const float* __restrict__ bK,
                              float* __restrict__ c) {
  int h = blockIdx.x, lane = threadIdx.x;
  float s = 0.f;
  for (int e = lane; e < DIM; e += 32)
    s = fmaf(q[h * DIM + e], bK[h * DIM + e], s);
  for (int off = 16; off; off >>= 1) s += __shfl_xor(s, off, 32);
  if (lane == 0) c[h] = s;
}

// ---- logits[h,m] = (qw[h,:]·K[m,:] + c[h]) / 1024 ; one wave per output ----
__global__ void logits_kernel(const float* __restrict__ qw,
                              const float* __restrict__ K,
                              const float* __restrict__ c,
                              float* __restrict__ logits) {
  int wid  = (blockIdx.x * blockDim.x + threadIdx.x) >> 5;  // h*2048+m
  int lane = threadIdx.x & 31;
  int h = wid >> 11;
  int m = wid & 2047;
  const float* kk = K + (size_t)m * DIM;
  const float* qh = qw + h * DIM;
  float s = 0.f;
  for (int d = lane; d < DIM; d += 32) s = fmaf(qh[d], kk[d], s);
  for (int off = 16; off; off >>= 1) s += __shfl_xor(s, off, 32);
  if (lane == 0) logits[wid] = (s + c[h]) * (1.0f / 1024.0f);
}

// ---- softmax over m; write wpad[16][2048] with rows 8..15 zeroed -----------
__global__ void softmax_kernel(const float* __restrict__ logits,
                               float* __restrict__ wpad) {
  __shared__ float red[256];
  int h = blockIdx.x;      // 0..7
  int t = threadIdx.x;     // 0..255
  const float* L = logits + h * MNB;
  float v[8], lmax = -INFINITY;
  for (int i = 0; i < 8; ++i) { v[i] = L[t + i * 256]; lmax = fmaxf(lmax, v[i]); }
  red[t] = lmax; __syncthreads();
  for (int s = 128; s; s >>= 1) { if (t < s) red[t] = fmaxf(red[t], red[t + s]); __syncthreads(); }
  float mx = red[0]; __syncthreads();
  float lsum = 0.f;
  for (int i = 0; i < 8; ++i) { v[i] = __expf(v[i] - mx); lsum += v[i]; }
  red[t] = lsum; __syncthreads();
  for (int s = 128; s; s >>= 1) { if (t < s) red[t] += red[t + s]; __syncthreads(); }
  float inv = 1.0f / red[0];
  for (int i = 0; i < 8; ++i) {
    wpad[h * MNB + t + i * 256]        = v[i] * inv;  // real head row
    wpad[(h + 8) * MNB + t + i * 256]  = 0.0f;        // zero pad row
  }
}

// ---- out = wpad @ V via V_WMMA_F32_16X16X4_F32 -----------------------------
// A (16x4): lane l -> wpad[l&15][k0 + 2*(l>>4) + {0,1}]   (aligned b64 load)
// B (4x16): lane l -> V[k0 + 2*(l>>4) + {0,1}][n0 + (l&15)]
// D (16x16): VGPR r, lanes 0..15 = (head r, pixel n0+lane)
// Each wave: 2 adjacent N-tiles (A reused), 4 accumulator chains.
// blockIdx.y selects K-partition; partials go to pbuf[part].
__global__ void out_wmma_split_kernel(const float* __restrict__ wpad,
                                      const float* __restrict__ V,
                                      float* __restrict__ pbuf) {
  const int n0   = blockIdx.x * 32;
  const int part = blockIdx.y;         // 0..KSPLIT-1
  const int kb   = part * KCHUNK;
  const int lane = threadIdx.x;        // 0..31, one wave per block
  const int col  = lane & 15;
  const int koff = (lane >> 4) * 2;    // 0 or 2

  const float* wrow  = wpad + col * MNB + kb + koff;
  const float* vcol0 = V + (size_t)(kb + koff) * NPIX + n0 + col;
  const float* vcol1 = vcol0 + 16;
  float* pout = pbuf + (size_t)part * (NH * NPIX);

  v8f a00 = {}, a01 = {}, a10 = {}, a11 = {};
#pragma unroll 2
  for (int k0 = 0; k0 < KCHUNK; k0 += 8) {
    v2f a0 = *(const v2f*)(wrow + k0);
    v2f b00, b10;
    b00.x = vcol0[(size_t)k0 * NPIX];
    b00.y = vcol0[(size_t)(k0 + 1) * NPIX];
    b10.x = vcol1[(size_t)k0 * NPIX];
    b10.y = vcol1[(size_t)(k0 + 1) * NPIX];
    a00 = __builtin_amdgcn_wmma_f32_16x16x4_f32(false, a0, false, b00,
                                                (short)0, a00, false, false);
    a10 = __builtin_amdgcn_wmma_f32_16x16x4_f32(false, a0, false, b10,
                                                (short)0, a10, false, false);
    v2f a1 = *(const v2f*)(wrow + k0 + 4);
    v2f b01, b11;
    b01.x = vcol0[(size_t)(k0 + 4) * NPIX];
    b01.y = vcol0[(size_t)(k0 + 5) * NPIX];
    b11.x = vcol1[(size_t)(k0 + 4) * NPIX];
    b11.y = vcol1[(size_t)(k0 + 5) * NPIX];
    a01 = __builtin_amdgcn_wmma_f32_16x16x4_f32(false, a1, false, b01,
                                                (short)0, a01, false, false);
    a11 = __builtin_amdgcn_wmma_f32_16x16x4_f32(false, a1, false, b11,
                                                (short)0, a11, false, false);
  }
  v8f t0 = a00 + a01;
  v8f t1 = a10 + a11;

  if (lane < 16) {
#pragma unroll
    for (int r = 0; r < NH; ++r) {
      pout[(size_t)r * NPIX + n0 + lane]      = t0[r];
      pout[(size_t)r * NPIX + n0 + 16 + lane] = t1[r];
    }
  }
}

// ---- combine K-partitions: out = pbuf[0] + pbuf[1] (vectorized) ------------
__global__ void combine_kernel(const float* __restrict__ pbuf,
                               float* __restrict__ out) {
  int i = blockIdx.x * blockDim.x + threadIdx.x;   // over NH*NPIX/4 float4s
  const v4f* p0 = (const v4f*)pbuf;
  const v4f* p1 = (const v4f*)(pbuf + (size_t)NH * NPIX);
  ((v4f*)out)[i] = p0[i] + p1[i];
}

// ---- fallback: single-partition, 1 tile/wave, writes d_out directly --------
__global__ void out_wmma_kernel(const float* __restrict__ wpad,
                                const float* __restrict__ V,
                                float* __restrict__ out) {
  const int n0   = blockIdx.x * 16;
  const int lane = threadIdx.x;
  const int col  = lane & 15;
  const int koff = (lane >> 4) * 2;

  const float* wrow = wpad + col * MNB + koff;
  const float* vcol = V + (size_t)koff * NPIX + n0 + col;

  v8f acc0 = {}, acc1 = {};
  for (int k0 = 0; k0 < MNB; k0 += 8) {
    v2f a0 = *(const v2f*)(wrow + k0);
    v2f b0;
    b0.x = vcol[(size_t)k0 * NPIX];
    b0.y = vcol[(size_t)(k0 + 1) * NPIX];
    acc0 = __builtin_amdgcn_wmma_f32_16x16x4_f32(false, a0, false, b0,
                                                 (short)0, acc0, false, false);
    v2f a1 = *(const v2f*)(wrow + k0 + 4);
    v2f b1;
    b1.x = vcol[(size_t)(k0 + 4) * NPIX];
    b1.y = vcol[(size_t)(k0 + 5) * NPIX];
    acc1 = __builtin_amdgcn_wmma_f32_16x16x4_f32(false, a1, false, b1,
                                                 (short)0, acc1, false, false);
  }
  v8f acc = acc0 + acc1;
  if (lane < 16) {
#pragma unroll
    for (int r = 0; r < NH; ++r)
      out[(size_t)r * NPIX + n0 + lane] = acc[r];
  }
}

// ---------------------------------------------------------------------------
extern "C" void kernel_launch(void* const* d_in, const int* in_sizes, int n_in,
                              void* d_out, int out_size, void* d_ws, size_t ws_size,
                              hipStream_t stream) {
  const float* Q  = (const float*)d_in[0];   // [1,1024]
  const float* K  = (const float*)d_in[1];   // [2048,1024]
  const float* V  = (const float*)d_in[2];   // [2048,128,128,3]
  const float* WQ = (const float*)d_in[3];   // [8,1024,1024]
  const float* bQ = (const float*)d_in[4];   // [8,1024]
  const float* WK = (const float*)d_in[5];   // [8,1024,1024]
  const float* bK = (const float*)d_in[6];   // [8,1024]
  float* out = (float*)d_out;                // [8,128,128,3]

  float* ws     = (float*)d_ws;
  float* q      = ws;              // 8192 floats
  float* qw     = ws + 8192;       // 8192 floats
  float* c      = ws + 16384;      // 16 floats (8 used)
  float* logits = ws + 16416;      // 16384 floats
  float* wpad   = ws + 32896;      // 32768 floats, 128B-aligned offset
  float* pbuf   = ws + 65664;      // KSPLIT * 8 * 49152 floats (partials)
  const size_t need = ((size_t)65664 + (size_t)KSPLIT * NH * NPIX) * sizeof(float);

  // q = Q·WQ^T + bQ      : 8192 waves
  proj_q_kernel<<<(NH * DIM * 32) / 256, 256, 0, stream>>>(Q, WQ, bQ, q);
  // qw = q·WK            : 8192 threads
  proj_qw_kernel<<<(NH * DIM) / 256, 256, 0, stream>>>(q, WK, qw);
  // c[h] = q[h]·bK[h]    : 8 waves
  proj_c_kernel<<<NH, 32, 0, stream>>>(q, bK, c);
  // logits               : 16384 waves
  logits_kernel<<<(NH * MNB * 32) / 256, 256, 0, stream>>>(qw, K, c, logits);
  // softmax + zero-pad to 16 rows
  softmax_kernel<<<NH, 256, 0, stream>>>(logits, wpad);

  if (ws_size >= need) {
    // out = wpad @ V : 2 tiles/wave, K-split(2) for HBM-saturating MLP
    dim3 grid(NPIX / 32, KSPLIT);
    out_wmma_split_kernel<<<grid, 32, 0, stream>>>(wpad, V, pbuf);
    combine_kernel<<<(NH * NPIX / 4) / 256, 256, 0, stream>>>(pbuf, out);
  } else {
    out_wmma_kernel<<<NPIX / 16, 32, 0, stream>>>(wpad, V, out);
  }
}